// GCNModel_57672820851285
// MI455X (gfx1250) — compile-verified
//
#include <hip/hip_runtime.h>
#include <math.h>

#define N_NODES 50000
#define N_EDGES 600000
#define D 128
#define N_LAYERS 3

typedef float v2f __attribute__((ext_vector_type(2)));
typedef float v8f __attribute__((ext_vector_type(8)));

// ---------------------------------------------------------------------------
// Degree / normalization
// ---------------------------------------------------------------------------
__global__ __launch_bounds__(256) void k_deg_init(float* __restrict__ deg, int n) {
    int i = blockIdx.x * blockDim.x + threadIdx.x;
    if (i < n) deg[i] = 1.0f;  // self-loop contributes 1
}

__global__ __launch_bounds__(256) void k_deg_scatter(const int* __restrict__ dst,
                                                     float* __restrict__ deg, int e) {
    int i = blockIdx.x * blockDim.x + threadIdx.x;
    if (i < e) atomicAdd(&deg[dst[i]], 1.0f);
}

__global__ __launch_bounds__(256) void k_inv_sqrt(float* __restrict__ deg, int n) {
    int i = blockIdx.x * blockDim.x + threadIdx.x;
    if (i < n) {
        float d = deg[i];
        deg[i] = (d > 0.0f) ? rsqrtf(d) : 0.0f;  // in place: deg -> inv_sqrt
    }
}

// ---------------------------------------------------------------------------
// GEMM: out[N x 128] = A[N x 128] @ W[128 x 128]  via V_WMMA_F32_16X16X4_F32
// Block = (32, 8): 8 waves, each wave owns one 16x16 output tile of a 16-row
// stripe (8 N-tiles cover D=128). K unrolled 32x (step 4).
// ---------------------------------------------------------------------------
__global__ __launch_bounds__(256) void k_gemm_wmma(const float* __restrict__ A,
                                                   const float* __restrict__ W,
                                                   float* __restrict__ out) {
    const int lane = threadIdx.x;       // 0..31
    const int wave = threadIdx.y;       // 0..7 -> N-tile
    const int m0   = blockIdx.x * 16;   // row stripe
    const int n0   = wave * 16;
    const int half = lane >> 4;         // 0: lanes 0-15, 1: lanes 16-31
    const int l16  = lane & 15;

    v8f acc = {0.f, 0.f, 0.f, 0.f, 0.f, 0.f, 0.f, 0.f};
    const float* arow = A + (size_t)(m0 + l16) * D;

#pragma unroll
    for (int k = 0; k < D; k += 4) {
        // A tile 16x4: VGPR0 = K=(half*2), VGPR1 = K=(half*2+1), M = l16
        v2f a = *(const v2f*)(arow + k + half * 2);
        // B tile 4x16: VGPRj holds rows K=j (lanes 0-15) and K=j+2 (lanes 16-31)
        const float* bp = W + (size_t)(k + half * 2) * D + n0 + l16;
        v2f b;
        b.x = bp[0];
        b.y = bp[D];
        acc = __builtin_amdgcn_wmma_f32_16x16x4_f32(
            /*neg_a=*/false, a, /*neg_b=*/false, b,
            /*c_mod=*/(short)0, acc, /*reuse_a=*/false, /*reuse_b=*/false);
    }

    // C/D layout: VGPR j -> (M=j, lanes 0-15) and (M=j+8, lanes 16-31), N=l16
#pragma unroll
    for (int j = 0; j < 8; ++j) {
        int row = m0 + j + half * 8;
        out[(size_t)row * D + n0 + l16] = acc[j];
    }
}

// ---------------------------------------------------------------------------
// Self-loop init: agg[i, :] = t[i, :] * inv_sqrt[i]^2   (also zero-initializes)
// One thread per float4.
// ---------------------------------------------------------------------------
__global__ __launch_bounds__(256) void k_selfloop(const float* __restrict__ t,
                                                  const float* __restrict__ isq,
                                                  float* __restrict__ agg) {
    int idx = blockIdx.x * blockDim.x + threadIdx.x;  // over N*D/4
    int row = idx >> 5;                               // 32 float4 per row
    if (row >= N_NODES) return;
    float s = isq[row];
    float nrm = s * s;
    float4 v = ((const float4*)t)[idx];
    v.x *= nrm; v.y *= nrm; v.z *= nrm; v.w *= nrm;
    ((float4*)agg)[idx] = v;
}

// ---------------------------------------------------------------------------
// Edge scatter: one wave (32 lanes) per edge, each lane does a float4 slice.
// agg[dst, :] += t[src, :] * isq[src]*isq[dst]
// ---------------------------------------------------------------------------
__global__ __launch_bounds__(256) void k_edge_scatter(const float* __restrict__ t,
                                                      const int* __restrict__ src,
                                                      const int* __restrict__ dst,
                                                      const float* __restrict__ isq,
                                                      float* __restrict__ agg, int e) {
    int warp = (blockIdx.x * blockDim.x + threadIdx.x) >> 5;
    int lane = threadIdx.x & 31;
    if (warp >= e) return;
    int s = src[warp];
    int d = dst[warp];
    float nrm = isq[s] * isq[d];
    float4 v = ((const float4*)(t + (size_t)s * D))[lane];
    float* ap = agg + (size_t)d * D + lane * 4;
    atomicAdd(ap + 0, v.x * nrm);
    atomicAdd(ap + 1, v.y * nrm);
    atomicAdd(ap + 2, v.z * nrm);
    atomicAdd(ap + 3, v.w * nrm);
}

// ---------------------------------------------------------------------------
// Bias + ReLU: h[i,:] = relu(agg[i,:] + b[:])
// ---------------------------------------------------------------------------
__global__ __launch_bounds__(256) void k_bias_relu(const float* __restrict__ agg,
                                                   const float* __restrict__ bias,
                                                   float* __restrict__ h) {
    int idx = blockIdx.x * blockDim.x + threadIdx.x;  // over N*D/4
    int row = idx >> 5;
    if (row >= N_NODES) return;
    float4 v = ((const float4*)agg)[idx];
    float4 b = ((const float4*)bias)[idx & 31];
    v.x = fmaxf(v.x + b.x, 0.f);
    v.y = fmaxf(v.y + b.y, 0.f);
    v.z = fmaxf(v.z + b.z, 0.f);
    v.w = fmaxf(v.w + b.w, 0.f);
    ((float4*)h)[idx] = v;
}

// ---------------------------------------------------------------------------
// Head: out[i] = sigmoid(dot(h[i,:], Wc) + bc)  -- one wave per node
// ---------------------------------------------------------------------------
__global__ __launch_bounds__(256) void k_head(const float* __restrict__ h,
                                              const float* __restrict__ Wc,
                                              const float* __restrict__ bc,
                                              float* __restrict__ out) {
    int node = (blockIdx.x * blockDim.x + threadIdx.x) >> 5;
    int lane = threadIdx.x & 31;
    if (node >= N_NODES) return;
    float4 hv = ((const float4*)(h + (size_t)node * D))[lane];
    float4 wv = ((const float4*)Wc)[lane];
    float s = hv.x * wv.x + hv.y * wv.y + hv.z * wv.z + hv.w * wv.w;
#pragma unroll
    for (int off = 16; off >= 1; off >>= 1)
        s += __shfl_xor(s, off, 32);
    if (lane == 0) out[node] = 1.0f / (1.0f + expf(-(s + bc[0])));
}

// ---------------------------------------------------------------------------
// Host-side launch sequence
// ---------------------------------------------------------------------------
extern "C" void kernel_launch(void* const* d_in, const int* in_sizes, int n_in,
                              void* d_out, int out_size, void* d_ws, size_t ws_size,
                              hipStream_t stream) {
    const float* x   = (const float*)d_in[0];           // [N, D]
    const int*   ei  = (const int*)d_in[1];             // [2, E] (src row, dst row)
    const float* Ws  = (const float*)d_in[2];           // [3, D, D]
    const float* bs  = (const float*)d_in[3];           // [3, D]
    const float* Wc  = (const float*)d_in[4];           // [D, 1]
    const float* bc  = (const float*)d_in[5];           // [1]
    float*       out = (float*)d_out;                   // [N, 1]

    const int* src = ei;
    const int* dst = ei + N_EDGES;

    const size_t nd = (size_t)N_NODES * D;
    float* isq = (float*)d_ws;                                  // N floats (deg -> inv_sqrt)
    float* t   = isq + ((N_NODES + 63) / 64) * 64;              // N*D (GEMM output)
    float* agg = t + nd;                                        // N*D (scatter accumulator)
    float* h   = agg + nd;                                      // N*D (layer activations)

    const int ND4   = (N_NODES * D) / 4;      // 1.6M float4 elements
    const int B     = 256;
    const int gN    = (N_NODES + B - 1) / B;
    const int gE    = (N_EDGES + B - 1) / B;
    const int gND4  = (ND4 + B - 1) / B;
    const int gEdgeWave = (N_EDGES * 32 + B - 1) / B;   // wave per edge
    const int gNodeWave = (N_NODES * 32 + B - 1) / B;   // wave per node

    // --- degree / normalization (isq doubles as deg) ---
    k_deg_init<<<gN, B, 0, stream>>>(isq, N_NODES);
    k_deg_scatter<<<gE, B, 0, stream>>>(dst, isq, N_EDGES);
    k_inv_sqrt<<<gN, B, 0, stream>>>(isq, N_NODES);

    // --- 3 GCN layers ---
    const float* hin = x;
    for (int layer = 0; layer < N_LAYERS; ++layer) {
        k_gemm_wmma<<<N_NODES / 16, dim3(32, 8), 0, stream>>>(hin, Ws + (size_t)layer * D * D, t);
        k_selfloop<<<gND4, B, 0, stream>>>(t, isq, agg);
        k_edge_scatter<<<gEdgeWave, B, 0, stream>>>(t, src, dst, isq, agg, N_EDGES);
        k_bias_relu<<<gND4, B, 0, stream>>>(agg, bs + (size_t)layer * D, h);
        hin = h;
    }

    // --- head ---
    k_head<<<gNodeWave, B, 0, stream>>>(h, Wc, bc, out);
}